// CenterNetHeatmapMaxDet_29480655520321
// MI455X (gfx1250) — compile-verified
//
#include <hip/hip_runtime.h>
#include <hip/hip_bf16.h>
#include <stdint.h>

// CenterNet top-K decode for x:(64,84,128,128) f32.
//   heatmap = ch[0:80), wh = ch[80:82), reg = ch[82:84)
// Phase 1: exact per-segment top-40 (segment = 8192 elems, 160 segs/batch),
//          staged through LDS with CDNA5 async global->LDS b128 copies.
// Phase 2: merge 160*40 candidates per batch, gather reg/wh, emit (64,40,6).
// Workspace requirement: 64*160*40*8 = 3,276,800 bytes.

#define NSEG 160          // segments per batch (8192 elems each; 160*8192 = 80*16384)
#define SEGE 8192         // elements per segment
#define TPB  256          // threads per block (8 wave32)
#define EPT  32           // elements per thread in phase 1
#define K    40
#define HW   16384        // 128*128
#define CSTRIDE 16384     // channel stride
#define BSTRIDE (84*16384)

// ---- monotonic float <-> sortable u32 ----
__device__ __forceinline__ unsigned f2s(float f) {
    unsigned u = __float_as_uint(f);
    return (u & 0x80000000u) ? ~u : (u | 0x80000000u);
}
__device__ __forceinline__ float s2f(unsigned s) {
    unsigned u = (s & 0x80000000u) ? (s & 0x7FFFFFFFu) : ~s;
    return __uint_as_float(u);
}

// ---- CDNA5 async global->LDS copy (16B per lane), tracked on ASYNCcnt ----
__device__ __forceinline__ void async_load_b128(unsigned lds_byte_addr, const void* gaddr) {
    asm volatile("global_load_async_to_lds_b128 %0, %1, off"
                 :
                 : "v"(lds_byte_addr), "v"(gaddr)
                 : "memory");
}
__device__ __forceinline__ void wait_asynccnt0() {
    asm volatile("s_wait_asynccnt 0" ::: "memory");
}

// ---- 64-bit max shuffle across wave32, then 8-wave LDS combine ----
__device__ __forceinline__ unsigned long long shfl_xor_u64(unsigned long long v, int m) {
    unsigned lo = (unsigned)v, hi = (unsigned)(v >> 32);
    lo = __shfl_xor(lo, m, 32);
    hi = __shfl_xor(hi, m, 32);
    return ((unsigned long long)hi << 32) | lo;
}
__device__ __forceinline__ unsigned long long
block_max_u64(unsigned long long key, unsigned long long* wred, int tid) {
    #pragma unroll
    for (int off = 16; off >= 1; off >>= 1) {
        unsigned long long o = shfl_xor_u64(key, off);
        if (o > key) key = o;
    }
    if ((tid & 31) == 0) wred[tid >> 5] = key;
    __syncthreads();
    unsigned long long best = wred[0];
    #pragma unroll
    for (int w = 1; w < 8; ++w) {
        unsigned long long o = wred[w];
        if (o > best) best = o;
    }
    __syncthreads();   // protect wred before next round
    return best;
}

// ================= Phase 1: per-segment exact top-40 =================
__global__ __launch_bounds__(TPB)
void topk_partial_kernel(const float* __restrict__ x,
                         unsigned long long* __restrict__ cand) {
    const int tid = threadIdx.x;
    const int s   = blockIdx.x;   // segment 0..159
    const int b   = blockIdx.y;   // batch   0..63

    __shared__ float4 tile[SEGE / 4];              // 32 KB
    __shared__ unsigned long long wred[8];

    const float* gbase = x + (long long)b * BSTRIDE + (long long)s * SEGE;

    // ---- stream 32 KB into LDS via async b128 copies (ASYNCcnt) ----
    #pragma unroll
    for (int j = 0; j < 8; ++j) {
        int t = j * TPB + tid;                     // float4 index in tile
        unsigned lds = (unsigned)(uintptr_t)(&tile[t]);
        async_load_b128(lds, (const void*)(gbase + (long long)t * 4));
    }
    wait_asynccnt0();
    __syncthreads();

    // ---- pull 32 elements/thread into registers as sortable keys ----
    unsigned sv[EPT];
    #pragma unroll
    for (int j = 0; j < 8; ++j) {
        float4 v = tile[j * TPB + tid];
        sv[4 * j + 0] = f2s(v.x);
        sv[4 * j + 1] = f2s(v.y);
        sv[4 * j + 2] = f2s(v.z);
        sv[4 * j + 3] = f2s(v.w);
    }
    // element p = 4*j+c of this thread has within-batch heatmap flat index:
    //   gidx = s*8192 + (p>>2)*1024 + tid*4 + (p&3)   (ascending in p)
    const unsigned segbase = (unsigned)s * SEGE + (unsigned)tid * 4;

    unsigned used = 0;
    unsigned bestv = 0;
    int bestp = -1;

    // lazy local argmax: rescan only on first round / after winning
    auto rescan = [&]() {
        bestp = -1; bestv = 0;
        #pragma unroll
        for (int p = 0; p < EPT; ++p) {
            if (!((used >> p) & 1u)) {
                unsigned v = sv[p];
                if (bestp < 0 || v > bestv) { bestv = v; bestp = p; }  // strict '>' keeps lowest idx on tie
            }
        }
    };
    rescan();

    unsigned long long* cblock = cand + ((long long)b * NSEG + s) * K;
    for (int k = 0; k < K; ++k) {
        unsigned long long key = 0;
        if (bestp >= 0) {
            unsigned gidx = segbase + (unsigned)(bestp >> 2) * 1024u + (unsigned)(bestp & 3);
            key = ((unsigned long long)bestv << 32) | (unsigned)(~gidx);  // ~idx: ties -> smaller idx wins
        }
        unsigned long long bmax = block_max_u64(key, wred, tid);
        if (tid == 0) cblock[k] = bmax;
        if (bestp >= 0 && key == bmax) {           // unique winner (idx packed in key)
            used |= 1u << bestp;
            rescan();
        }
    }
}

// ================= Phase 2: merge + gather + bbox =================
__global__ __launch_bounds__(TPB)
void topk_final_kernel(const float* __restrict__ x,
                       const unsigned long long* __restrict__ cand,
                       float* __restrict__ out) {
    const int tid = threadIdx.x;
    const int b   = blockIdx.x;
    const int CPT = (NSEG * K) / TPB;              // 25 candidates per thread

    __shared__ unsigned long long wred[8];
    __shared__ unsigned long long winners[K];

    const unsigned long long* cb = cand + (long long)b * (NSEG * K);
    unsigned long long kreg[CPT];
    #pragma unroll
    for (int i = 0; i < CPT; ++i) kreg[i] = cb[i * TPB + tid];

    unsigned used = 0;
    unsigned long long bestk = 0;
    int bestp = -1;
    auto rescan = [&]() {
        bestp = -1; bestk = 0;
        #pragma unroll
        for (int i = 0; i < CPT; ++i) {
            if (!((used >> i) & 1u)) {
                unsigned long long v = kreg[i];
                if (bestp < 0 || v > bestk) { bestk = v; bestp = i; }
            }
        }
    };
    rescan();

    for (int k = 0; k < K; ++k) {
        unsigned long long key = (bestp >= 0) ? bestk : 0ull;
        unsigned long long bmax = block_max_u64(key, wred, tid);
        if (tid == 0) winners[k] = bmax;
        if (bestp >= 0 && key == bmax) {
            used |= 1u << bestp;
            rescan();
        }
    }
    __syncthreads();

    if (tid < K) {
        unsigned long long key = winners[tid];
        float    score = s2f((unsigned)(key >> 32));
        unsigned gidx  = ~(unsigned)key;           // flat index into 80*16384
        unsigned cls   = gidx >> 14;
        unsigned spat  = gidx & (HW - 1);
        float ys = (float)(spat >> 7);
        float xs = (float)(spat & 127);

        const float* bb = x + (long long)b * BSTRIDE;
        float w  = bb[80 * CSTRIDE + spat];
        float h  = bb[81 * CSTRIDE + spat];
        float rx = bb[82 * CSTRIDE + spat];
        float ry = bb[83 * CSTRIDE + spat];

        float cx = xs + rx, cy = ys + ry;
        float hw = 0.5f * w, hh = 0.5f * h;

        float* o = out + ((long long)b * K + tid) * 6;
        o[0] = (cx - hw) * 4.0f;
        o[1] = (cy - hh) * 4.0f;
        o[2] = (cx + hw) * 4.0f;
        o[3] = (cy + hh) * 4.0f;
        o[4] = (float)cls;
        o[5] = score;
    }
}

extern "C" void kernel_launch(void* const* d_in, const int* in_sizes, int n_in,
                              void* d_out, int out_size, void* d_ws, size_t ws_size,
                              hipStream_t stream) {
    (void)in_sizes; (void)n_in; (void)out_size; (void)ws_size;
    const float* x = (const float*)d_in[0];
    float* out = (float*)d_out;
    unsigned long long* cand = (unsigned long long*)d_ws;  // needs 3,276,800 bytes

    topk_partial_kernel<<<dim3(NSEG, 64), TPB, 0, stream>>>(x, cand);
    topk_final_kernel<<<64, TPB, 0, stream>>>(x, cand, out);
}